// SPDTangentProjection_56616258896176
// MI455X (gfx1250) — compile-verified
//
#include <hip/hip_runtime.h>
#include <hip/hip_bf16.h>

// SPD eigenvalue clamp + matrix logarithm, matmul-only formulation for CDNA5.
// All heavy math is fp32 WMMA (v_wmma_f32_16x16x4_f32), LDS-resident matrices.
// Newton-Schulz sign -> spectral clamp; coupled Newton-Schulz sqrt chain +
// Taylor/Horner series -> matrix log. All updates fused into GEMM epilogues.

typedef float v2f __attribute__((ext_vector_type(2)));
typedef float v8f __attribute__((ext_vector_type(8)));

#define MDIM      64
#define LDSTR     65          // padded row stride (floats) -> conflict-free LDS
#define NTHREADS  512         // 16 wave32 waves: one 16x16 tile per wave
#define EPSV      1e-3f

#define SIGN_ITERS 28         // Newton-Schulz matrix sign (even!)
#define NUM_SQRTS  5          // inverse scaling-and-squaring depth (2^5)
#define SQRT_ITERS 14         // coupled Newton-Schulz sqrt iterations (even!)
#define LOG_TERMS  16         // Taylor terms for log(I - E)

static_assert(SIGN_ITERS % 2 == 0, "sign ping-pong must end in M0");
static_assert(SQRT_ITERS % 2 == 0, "sqrt ping-pong must restore Y->M0, Z->M1");

// ---------------------------------------------------------------------------
// 64x64 fp32 GEMM on LDS, whole workgroup (16 waves, EXEC all-ones).
//   C = alpha*(A*B) + beta*C0 (optional) + gamma*I
// C must not alias A or B (C0 may equal A or B source-side; it is only read).
// Ends with a block barrier.
// WMMA f32 16x16x4 operand layout (ISA 7.12.2, 32-bit):
//   A (16x4): lanes 0-15 row M=l hold K={k0,k0+1}; lanes 16-31 K={k0+2,k0+3}
//   B (4x16): lanes hold column N=l; VGPR pair holds two K rows per half-wave
//   C/D (16x16): VGPR r -> row r (lanes 0-15) / row r+8 (lanes 16-31)
// ---------------------------------------------------------------------------
__device__ __forceinline__ void gemm64(float* __restrict__ C,
                                       const float* __restrict__ A,
                                       const float* __restrict__ B,
                                       float alpha,
                                       const float* __restrict__ C0,
                                       float beta, float gamma)
{
    const int lane = threadIdx.x & 31;
    const int wave = threadIdx.x >> 5;
    const int ti   = (wave >> 2) << 4;   // tile row base
    const int tj   = (wave & 3)  << 4;   // tile col base
    const int half = lane >> 4;          // 0: lanes 0-15, 1: lanes 16-31
    const int l    = lane & 15;
    const int arow = ti + l;
    const int bcol = tj + l;

    v8f acc = {};
#pragma unroll
    for (int kk = 0; kk < 16; ++kk) {
        const int k0 = (kk << 2) + (half << 1);
        v2f a, b;
        a.x = A[arow * LDSTR + k0];
        a.y = A[arow * LDSTR + k0 + 1];
        b.x = B[(k0)     * LDSTR + bcol];
        b.y = B[(k0 + 1) * LDSTR + bcol];
        // (neg_a, A, neg_b, B, c_mod, C, reuse_a, reuse_b)
        acc = __builtin_amdgcn_wmma_f32_16x16x4_f32(false, a, false, b,
                                                    (short)0, acc, false, false);
    }
#pragma unroll
    for (int r = 0; r < 8; ++r) {
        const int row = ti + r + (half << 3);
        const int col = tj + l;
        float v = alpha * acc[r];
        if (C0)         v += beta * C0[row * LDSTR + col];
        if (row == col) v += gamma;
        C[row * LDSTR + col] = v;
    }
    __syncthreads();
}

// D = a*S1 + g*I   (S1 may be nullptr; in-place allowed)
__device__ __forceinline__ void ew64(float* __restrict__ D,
                                     float a, const float* S1, float g)
{
    for (int e = threadIdx.x; e < MDIM * MDIM; e += NTHREADS) {
        const int i = e >> 6, j = e & 63;
        float v = 0.0f;
        if (S1)     v += a * S1[i * LDSTR + j];
        if (i == j) v += g;
        D[i * LDSTR + j] = v;
    }
    __syncthreads();
}

// squared Frobenius norm of an LDS matrix (block-wide tree reduction)
__device__ __forceinline__ float fro2(const float* M, float* red)
{
    float s = 0.0f;
    for (int e = threadIdx.x; e < MDIM * MDIM; e += NTHREADS) {
        const float v = M[(e >> 6) * LDSTR + (e & 63)];
        s += v * v;
    }
    red[threadIdx.x] = s;
    __syncthreads();
#pragma unroll
    for (int off = NTHREADS / 2; off > 0; off >>= 1) {
        if ((int)threadIdx.x < off) red[threadIdx.x] += red[threadIdx.x + off];
        __syncthreads();
    }
    const float r = red[0];
    __syncthreads();
    return r;
}

__global__ __launch_bounds__(NTHREADS, 1)
void spd_matrix_log_kernel(const float* __restrict__ Xg, float* __restrict__ Og)
{
    __shared__ float M0[MDIM * LDSTR];
    __shared__ float M1[MDIM * LDSTR];
    __shared__ float M2[MDIM * LDSTR];
    __shared__ float M3[MDIM * LDSTR];
    __shared__ float M4[MDIM * LDSTR];
    __shared__ float M5[MDIM * LDSTR];
    __shared__ float red[NTHREADS];

    const float* Xb = Xg + (size_t)blockIdx.x * (MDIM * MDIM);
    float*       Ob = Og + (size_t)blockIdx.x * (MDIM * MDIM);

    // ---- load X (coalesced, 8 dwords/thread) into padded LDS -> M4 ----
    for (int e = threadIdx.x; e < MDIM * MDIM; e += NTHREADS)
        M4[(e >> 6) * LDSTR + (e & 63)] = Xb[e];
    __syncthreads();

    // ---- sign(X) via Newton-Schulz: S <- 1.5 S - 0.5 S^3 (fused epilogue) ----
    const float c0 = sqrtf(fro2(M4, red)) + 1e-30f;   // spectral radius bound
    ew64(M0, 1.0f / c0, M4, 0.0f);                    // S = X / c0
    {
        float* s = M0; float* sn = M2;
#pragma unroll 1
        for (int it = 0; it < SIGN_ITERS; ++it) {
            gemm64(M1, s, s, 1.0f, nullptr, 0.0f, 0.0f);   // T  = S*S
            gemm64(sn, s, M1, -0.5f, s, 1.5f, 0.0f);       // S' = 1.5S - 0.5 S*T
            float* t = s; s = sn; sn = t;                  // even count -> ends M0
        }
    }

    // ---- eigenvalue clamp: P = (I+S)/2 ; Xspd = P X P + eps (I - P) ----
    ew64(M1, 0.5f, M0, 0.5f);                              // P -> M1
    gemm64(M2, M1, M4, 1.0f, nullptr, 0.0f, 0.0f);         // T1 = P*X
    gemm64(M3, M2, M1, 1.0f, M1, -EPSV, EPSV);             // Xspd -> M3

    // ---- normalize so eigenvalues lie in (0,1] ----
    const float c1 = sqrtf(fro2(M3, red)) + 1e-30f;
    ew64(M0, 1.0f / c1, M3, 0.0f);                         // A = Y -> M0

    // ---- repeated square roots: coupled Newton-Schulz, fully GEMM-fused ----
    //   M  = Z*Y
    //   Y' = Y*W = 1.5*Y - 0.5*(Y*M)     (W = 1.5I - 0.5M never materialized)
    //   Z' = W*Z = 1.5*Z - 0.5*(M*Z)
#pragma unroll 1
    for (int lvl = 0; lvl < NUM_SQRTS; ++lvl) {
        float* yp = M0; float* yn = M3;                    // Y ping-pong
        float* zp = M1; float* zn = M5;                    // Z ping-pong
        ew64(zp, 0.0f, nullptr, 1.0f);                     // Z = I
#pragma unroll 1
        for (int it = 0; it < SQRT_ITERS; ++it) {
            gemm64(M2, zp, yp, 1.0f, nullptr, 0.0f, 0.0f); // M  = Z*Y
            gemm64(yn, yp, M2, -0.5f, yp, 1.5f, 0.0f);     // Y' = 1.5Y - 0.5 Y*M
            gemm64(zn, M2, zp, -0.5f, zp, 1.5f, 0.0f);     // Z' = 1.5Z - 0.5 M*Z
            float* t;
            t = yp; yp = yn; yn = t;
            t = zp; zp = zn; zn = t;
        }
        // even SQRT_ITERS -> Y back in M0 (= A for the next level), Z in M1
    }

    // ---- log(R) = -sum E^n/n,  E = I - R,  via Horner (all GEMMs) ----
    ew64(M2, -1.0f, M0, 1.0f);                             // E -> M2
    ew64(M3, 0.0f, nullptr, 1.0f / (float)LOG_TERMS);      // U = (1/m) I
    {
        float* U = M3; float* Un = M5;
#pragma unroll 1
        for (int j = LOG_TERMS - 1; j >= 1; --j) {
            gemm64(Un, M2, U, 1.0f, nullptr, 0.0f, 1.0f / (float)j); // U<-E*U+(1/j)I
            float* t = U; U = Un; Un = t;
        }
        gemm64(M4, M2, U, 1.0f, nullptr, 0.0f, 0.0f);      // Ssum = E*U = -log(R)
    }

    // ---- out = 2^k * log(R) + log(c1) * I ----
    const float SCALE = (float)(1 << NUM_SQRTS);
    const float lc    = logf(c1);
    for (int e = threadIdx.x; e < MDIM * MDIM; e += NTHREADS) {
        const int i = e >> 6, j = e & 63;
        float v = -SCALE * M4[i * LDSTR + j];
        if (i == j) v += lc;
        Ob[e] = v;
    }
}

extern "C" void kernel_launch(void* const* d_in, const int* in_sizes, int n_in,
                              void* d_out, int out_size, void* d_ws, size_t ws_size,
                              hipStream_t stream) {
    (void)n_in; (void)d_ws; (void)ws_size; (void)out_size;
    const float* x   = (const float*)d_in[0];
    float*       out = (float*)d_out;
    const int nmat = in_sizes[0] / (MDIM * MDIM);   // 4096 matrices
    spd_matrix_log_kernel<<<nmat, NTHREADS, 0, stream>>>(x, out);
}